// PKGNN_v1_13099650253150
// MI455X (gfx1250) — compile-verified
//
#include <hip/hip_runtime.h>
#include <hip/hip_bf16.h>

typedef float v2f __attribute__((ext_vector_type(2)));
typedef float v8f __attribute__((ext_vector_type(8)));
typedef unsigned int u32x4 __attribute__((ext_vector_type(4)));
typedef int i32x4 __attribute__((ext_vector_type(4)));
typedef int i32x8 __attribute__((ext_vector_type(8)));

#define B_ 16
#define N_ 4096
#define E_ 131072
#define H_ 16
#define T_ 3
#define FC_ 20
#define BH 256              // B_*H_  (per-node payload)
#define BC 48               // B_*3
#define INV_SQRT_DEG 0.17677669529663687f   // 1/sqrt(E/N) = 1/sqrt(32)

// workspace layout (float offsets)
#define OFF_EW    0                       // 8:  ew[l*4+t]
#define OFF_FINV  8                       // 144: Finv[b][3][3]
#define OFF_G     160                     // 768: G[b][j][k]
#define OFF_WLS   928                     // 512: WlS[l][16][16] (inv_sqrt_deg folded)
#define OFF_H     4096                    // N*256
#define OFF_AGG   (OFF_H   + N_*BH)       // N*256
#define OFF_HOUT  (OFF_AGG + N_*BH)       // N*48
#define OFF_AMAT  (OFF_HOUT + N_*BC)      // N*48
#define OFF_CMAT  (OFF_AMAT + N_*BC)      // N*48
#define OFF_T     (OFF_CMAT + N_*BC)      // N*48

// ---------------------------------------------------------------- TDM helper
// 1-D contiguous tile: nelem fp32 elements, global -> LDS (D# per ISA 08 §8).
__device__ __forceinline__ void tdm_load_1d(unsigned lds_off, const float* gptr, int nelem) {
  unsigned long long ga = (unsigned long long)(size_t)gptr;
  u32x4 g0;
  g0.x = 1u;                                            // count=1, user mode
  g0.y = lds_off;                                       // lds_addr (bytes)
  g0.z = (unsigned)(ga & 0xFFFFFFFFu);                  // global_addr[31:0]
  g0.w = (unsigned)((ga >> 32) & 0x1FFFFFFu) | (2u << 30); // addr[56:32] | type=2
  i32x8 g1;
  g1[0] = 2 << 16;                                      // data_size = 4B
  g1[1] = (nelem & 0xFFFF) << 16;                       // tensor_dim0[15:0]
  g1[2] = ((nelem >> 16) & 0xFFFF) | (1 << 16);         // tensor_dim0[31:16], tensor_dim1=1
  g1[3] = (nelem & 0xFFFF) << 16;                       // tile_dim0
  g1[4] = 0;                                            // tile_dim1=0 (1-D), tile_dim2=0
  g1[5] = nelem;                                        // tensor_dim0_stride[31:0]
  g1[6] = 0;
  g1[7] = 0;
  i32x4 gz4 = {0, 0, 0, 0};
  i32x8 gz8 = {0, 0, 0, 0, 0, 0, 0, 0};
  __builtin_amdgcn_tensor_load_to_lds(g0, g1, gz4, gz4, gz8, 0);
}

// ---------------------------------------------------------------- tiny precompute
__global__ void k_prep(const float* __restrict__ F, const float* __restrict__ W_in,
                       const float* __restrict__ fc1_0, const float* __restrict__ fc2_0,
                       const float* __restrict__ W_0,
                       const float* __restrict__ fc1_1, const float* __restrict__ fc2_1,
                       const float* __restrict__ W_1,
                       float* __restrict__ ws) {
  int tid = threadIdx.x;
  // per-type edge weights: ew_l[t] = relu(fc1_l[t,:]) . fc2_l
  if (tid < 6) {
    int l = tid / 3, t = tid % 3;
    const float* fc1 = l ? fc1_1 : fc1_0;
    const float* fc2 = l ? fc2_1 : fc2_0;
    float s = 0.f;
    for (int k = 0; k < FC_; ++k) {
      float v = fc1[t * FC_ + k];
      s += (v > 0.f ? v : 0.f) * fc2[k];
    }
    ws[OFF_EW + l * 4 + t] = s;
  }
  // 3x3 inverse of F per batch (adjugate / det)
  if (tid >= 32 && tid < 48) {
    int bb = tid - 32;
    const float* f = F + bb * 9;
    float a = f[0], b = f[1], c = f[2];
    float d = f[3], e = f[4], fe = f[5];
    float g = f[6], h = f[7], i = f[8];
    float A0 = e * i - fe * h;
    float A1 = -(d * i - fe * g);
    float A2 = d * h - e * g;
    float det = a * A0 + b * A1 + c * A2;
    float r = 1.f / det;
    float* inv = ws + OFF_FINV + bb * 9;
    inv[0] = A0 * r;
    inv[1] = -(b * i - c * h) * r;
    inv[2] = (b * fe - c * e) * r;
    inv[3] = A1 * r;
    inv[4] = (a * i - c * g) * r;
    inv[5] = -(a * fe - c * d) * r;
    inv[6] = A2 * r;
    inv[7] = -(a * h - b * g) * r;
    inv[8] = (a * e - b * d) * r;
  }
  // G[b][j][k] = sum_i F[b][i][j] * W_in[i][k]   (so h0 = pos . G)
  for (int idx = tid; idx < B_ * 3 * H_; idx += blockDim.x) {
    int b = idx / (3 * H_), j = (idx / H_) % 3, k = idx % H_;
    float s = 0.f;
    for (int i = 0; i < 3; ++i) s += F[b * 9 + i * 3 + j] * W_in[i * H_ + k];
    ws[OFF_G + idx] = s;
  }
  // scaled layer weights: WlS = W_l * inv_sqrt_deg
  for (int idx = tid; idx < 2 * H_ * H_; idx += blockDim.x) {
    const float* W = (idx < H_ * H_) ? W_0 : W_1;
    ws[OFF_WLS + idx] = W[idx & (H_ * H_ - 1)] * INV_SQRT_DEG;
  }
}

// ---------------------------------------------------------------- h0 = (F@pos)@W_in
__global__ void k_h0(const float* __restrict__ pos, const float* __restrict__ G,
                     float* __restrict__ h) {
  int n = blockIdx.x;
  int tid = threadIdx.x;        // = b*16 + k
  int b = tid >> 4, k = tid & 15;
  float p0 = pos[n * 3 + 0], p1 = pos[n * 3 + 1], p2 = pos[n * 3 + 2];
  const float* Gp = G + b * 48 + k;   // stride 16 over j
  h[n * BH + tid] = p0 * Gp[0] + p1 * Gp[16] + p2 * Gp[32];
}

// ---------------------------------------------------------------- zero fill
__global__ void k_zero(float4* __restrict__ p, int n4) {
  int i = blockIdx.x * blockDim.x + threadIdx.x;
  if (i < n4) p[i] = make_float4(0.f, 0.f, 0.f, 0.f);
}

// ---------------------------------------------------------------- edge scatter-add
__global__ void k_scatter(const float* __restrict__ h, float* __restrict__ agg,
                          const int* __restrict__ src, const int* __restrict__ dst,
                          const int* __restrict__ typ, const float* __restrict__ ew) {
  long g = (long)blockIdx.x * blockDim.x + threadIdx.x;
  int e  = (int)(g >> 6);            // 64 threads per edge
  int c4 = ((int)g & 63) << 2;       // 4 floats per thread
  float wgt = ew[typ[e]];
  const float4 v = *(const float4*)(h + (long)src[e] * BH + c4);
  float* a = agg + (long)dst[e] * BH + c4;
  unsafeAtomicAdd(a + 0, v.x * wgt);
  unsafeAtomicAdd(a + 1, v.y * wgt);
  unsafeAtomicAdd(a + 2, v.z * wgt);
  unsafeAtomicAdd(a + 3, v.w * wgt);
}

// ---------------------------------------------------------------- h = relu(agg @ WlS), per-batch 16x16 tiles via WMMA f32 16x16x4
__global__ void k_layer_gemm(const float* __restrict__ agg, float* __restrict__ h,
                             const float* __restrict__ WlS) {
  int wave = (int)((blockIdx.x * blockDim.x + threadIdx.x) >> 5);
  int lane = threadIdx.x & 31;
  int b  = wave >> 8;                 // batch 0..15
  int n0 = (wave & 255) << 4;         // node-tile base
  int row = lane & 15;
  int hi  = lane >> 4;
  v8f acc = {0.f, 0.f, 0.f, 0.f, 0.f, 0.f, 0.f, 0.f};
  #pragma unroll
  for (int kk0 = 0; kk0 < H_; kk0 += 4) {
    int k0 = kk0 + 2 * hi;
    const float* ap = agg + (n0 + row) * BH + b * H_ + k0;
    v2f A;  A.x  = ap[0];                 A.y  = ap[1];                 // A[m][k0], A[m][k0+1]
    v2f Bf; Bf.x = WlS[k0 * H_ + row];    Bf.y = WlS[(k0 + 1) * H_ + row]; // B[k][col]
    acc = __builtin_amdgcn_wmma_f32_16x16x4_f32(false, A, false, Bf,
                                                (short)0, acc, false, false);
  }
  int mbase = 8 * hi;
  int col = row;
  #pragma unroll
  for (int r = 0; r < 8; ++r) {
    float v = acc[r];
    h[(n0 + mbase + r) * BH + b * H_ + col] = v > 0.f ? v : 0.f;
  }
}

// ---------------------------------------------------------------- hout = h @ W_out
__global__ void k_wout(const float* __restrict__ h, const float* __restrict__ Wout,
                       float* __restrict__ hout) {
  int g = blockIdx.x * blockDim.x + threadIdx.x;   // n*48 + col
  int n = g / BC, col = g % BC;
  int b = col / 3, p = col % 3;
  const float* hr = h + n * BH + b * H_;
  float s = 0.f;
  #pragma unroll
  for (int k = 0; k < H_; ++k) s += hr[k] * Wout[k * 3 + p];
  hout[g] = s;
}

// ---------------------------------------------------------------- a,c for first N edges
__global__ void k_edge_ac(const float* __restrict__ hout, const int* __restrict__ src,
                          const int* __restrict__ dst, const float* __restrict__ Finv,
                          float* __restrict__ aM, float* __restrict__ cM) {
  int g = blockIdx.x * blockDim.x + threadIdx.x;   // e*16 + b, e < N_
  int e = g >> 4, b = g & 15;
  int s = src[e], d = dst[e];
  float av[3];
  #pragma unroll
  for (int p = 0; p < 3; ++p) {
    av[p] = hout[s * BC + b * 3 + p] - hout[d * BC + b * 3 + p];
    aM[e * BC + b * 3 + p] = av[p];
  }
  const float* fi = Finv + b * 9;
  #pragma unroll
  for (int q = 0; q < 3; ++q)
    cM[e * BC + b * 3 + q] =
        fi[q * 3 + 0] * av[0] + fi[q * 3 + 1] * av[1] + fi[q * 3 + 2] * av[2];
}

// ---------------------------------------------------------------- t = W2^T @ aM  (4096x4096 @ 4096x48)
// Block = 8 waves sharing one K chunk (512 rows of aM), each wave owns one jtile.
// aM slab staged into LDS by the Tensor Data Mover, double buffered (8 x 64-row chunks).
__global__ void k_big_gemm(const float* __restrict__ W2, const float* __restrict__ aM,
                           float* __restrict__ t) {
  __shared__ float sbuf[2][64 * BC];          // 2 x 12KB
  int wid  = threadIdx.x >> 5;                // wave in block: 0..7
  int lane = threadIdx.x & 31;
  int kc = blockIdx.x >> 5;                   // K chunk 0..7 (512 rows each)
  int jt = ((blockIdx.x & 31) << 3) + wid;    // jtile 0..255
  int j0 = jt << 4;
  int kbase = kc << 9;
  int row = lane & 15;
  int hi  = lane >> 4;
  v8f acc0 = {0.f, 0.f, 0.f, 0.f, 0.f, 0.f, 0.f, 0.f};
  v8f acc1 = {0.f, 0.f, 0.f, 0.f, 0.f, 0.f, 0.f, 0.f};
  v8f acc2 = {0.f, 0.f, 0.f, 0.f, 0.f, 0.f, 0.f, 0.f};

  // prologue: wave 0 DMAs chunks 0 and 1
  if (threadIdx.x < 32) {
    tdm_load_1d((unsigned)(size_t)&sbuf[0][0], aM + (size_t)kbase * BC, 64 * BC);
    tdm_load_1d((unsigned)(size_t)&sbuf[1][0], aM + (size_t)(kbase + 64) * BC, 64 * BC);
  }
  for (int c = 0; c < 8; ++c) {
    if (threadIdx.x < 32) {
      if (c == 7) __builtin_amdgcn_s_wait_tensorcnt(0);
      else        __builtin_amdgcn_s_wait_tensorcnt(1);
    }
    __syncthreads();                          // chunk c resident in sbuf[c&1]
    const float* lb = &sbuf[c & 1][0];
    int kg = kbase + (c << 6);
    #pragma unroll 4
    for (int kloc = 0; kloc < 64; kloc += 4) {
      int k0 = kloc + 2 * hi;
      long base = (long)(kg + k0) * N_ + j0 + row;   // W2[k][j0+row] (A = W2^T tile)
      v2f A;  A.x = W2[base];  A.y = W2[base + N_];
      const float* br = lb + k0 * BC + row;          // LDS: aM[k][ct*16+col]
      v2f B0; B0.x = br[0];  B0.y = br[BC];
      v2f B1; B1.x = br[16]; B1.y = br[16 + BC];
      v2f B2; B2.x = br[32]; B2.y = br[32 + BC];
      acc0 = __builtin_amdgcn_wmma_f32_16x16x4_f32(false, A, false, B0, (short)0, acc0, false, false);
      acc1 = __builtin_amdgcn_wmma_f32_16x16x4_f32(false, A, false, B1, (short)0, acc1, false, false);
      acc2 = __builtin_amdgcn_wmma_f32_16x16x4_f32(false, A, false, B2, (short)0, acc2, false, false);
    }
    __syncthreads();                          // everyone done with sbuf[c&1]
    if (threadIdx.x < 32 && c + 2 < 8) {
      tdm_load_1d((unsigned)(size_t)&sbuf[c & 1][0],
                  aM + (size_t)(kbase + ((c + 2) << 6)) * BC, 64 * BC);
    }
  }
  int mbase = 8 * hi;
  int col = row;
  #pragma unroll
  for (int r = 0; r < 8; ++r) {
    int j = j0 + mbase + r;
    unsafeAtomicAdd(&t[j * BC + col],      acc0[r]);
    unsafeAtomicAdd(&t[j * BC + 16 + col], acc1[r]);
    unsafeAtomicAdd(&t[j * BC + 32 + col], acc2[r]);
  }
}

// ---------------------------------------------------------------- stress[b,p,q] = sum_j t[j,b,p]*c[j,b,q]
__global__ void k_stress(const float* __restrict__ t, const float* __restrict__ cM,
                         float* __restrict__ out) {
  __shared__ float red[256];
  int b = blockIdx.x, tid = threadIdx.x;
  float acc[9];
  #pragma unroll
  for (int m = 0; m < 9; ++m) acc[m] = 0.f;
  for (int j = tid; j < N_; j += 256) {
    float tp[3], cq[3];
    #pragma unroll
    for (int p = 0; p < 3; ++p) tp[p] = t[j * BC + b * 3 + p];
    #pragma unroll
    for (int q = 0; q < 3; ++q) cq[q] = cM[j * BC + b * 3 + q];
    #pragma unroll
    for (int p = 0; p < 3; ++p)
      #pragma unroll
      for (int q = 0; q < 3; ++q)
        acc[p * 3 + q] += tp[p] * cq[q];
  }
  for (int m = 0; m < 9; ++m) {
    red[tid] = acc[m];
    __syncthreads();
    for (int s = 128; s > 0; s >>= 1) {
      if (tid < s) red[tid] += red[tid + s];
      __syncthreads();
    }
    if (tid == 0) out[b * 9 + m] = red[0];
    __syncthreads();
  }
}

// ---------------------------------------------------------------- launch
extern "C" void kernel_launch(void* const* d_in, const int* in_sizes, int n_in,
                              void* d_out, int out_size, void* d_ws, size_t ws_size,
                              hipStream_t stream) {
  const float* F     = (const float*)d_in[0];
  const float* pos   = (const float*)d_in[1];
  const int*   esrc  = (const int*)d_in[2];
  const int*   edst  = (const int*)d_in[3];
  const int*   etyp  = (const int*)d_in[4];
  const float* W_in  = (const float*)d_in[5];
  const float* fc1_0 = (const float*)d_in[6];
  const float* fc2_0 = (const float*)d_in[7];
  const float* W_0   = (const float*)d_in[8];
  const float* fc1_1 = (const float*)d_in[9];
  const float* fc2_1 = (const float*)d_in[10];
  const float* W_1   = (const float*)d_in[11];
  const float* W_out = (const float*)d_in[12];
  const float* w     = (const float*)d_in[13];
  float* ws  = (float*)d_ws;
  float* out = (float*)d_out;

  k_prep<<<1, 256, 0, stream>>>(F, W_in, fc1_0, fc2_0, W_0, fc1_1, fc2_1, W_1, ws);
  k_h0<<<N_, 256, 0, stream>>>(pos, ws + OFF_G, ws + OFF_H);
  for (int l = 0; l < 2; ++l) {
    k_zero<<<(N_ * BH / 4) / 256, 256, 0, stream>>>((float4*)(ws + OFF_AGG), N_ * BH / 4);
    k_scatter<<<(E_ * 64) / 256, 256, 0, stream>>>(ws + OFF_H, ws + OFF_AGG,
                                                   esrc, edst, etyp, ws + OFF_EW + l * 4);
    k_layer_gemm<<<(B_ * (N_ / 16)) / 8, 256, 0, stream>>>(ws + OFF_AGG, ws + OFF_H,
                                                           ws + OFF_WLS + l * 256);
  }
  k_wout<<<(N_ * BC) / 256, 256, 0, stream>>>(ws + OFF_H, W_out, ws + OFF_HOUT);
  k_edge_ac<<<(N_ * 16) / 256, 256, 0, stream>>>(ws + OFF_HOUT, esrc, edst,
                                                 ws + OFF_FINV, ws + OFF_AMAT, ws + OFF_CMAT);
  k_zero<<<(N_ * BC / 4 + 255) / 256, 256, 0, stream>>>((float4*)(ws + OFF_T), N_ * BC / 4);
  k_big_gemm<<<256, 256, 0, stream>>>(w, ws + OFF_AMAT, ws + OFF_T);
  k_stress<<<B_, 256, 0, stream>>>(ws + OFF_T, ws + OFF_CMAT, out);
}